// ParticleNetSimple_55972013802030
// MI455X (gfx1250) — compile-verified
//
#include <hip/hip_runtime.h>

typedef __attribute__((ext_vector_type(2))) float v2f;
typedef __attribute__((ext_vector_type(8))) float v8f;

#define DCH 64
#define NPTS 256
#define NBATCH 32
#define NROWS (NBATCH * NPTS) /* 8192 */
#define BSTRIDE 144 /* LDS row pitch (floats): 2*144 % 64 = 32 -> halves hit disjoint banks */

// ---------------------------------------------------------------------------
// Kernel 1: h0 = bn(relu(x @ W1 + b1))   x:(8192,4) W1:(4,64) -> h0:(8192,64)
// One wave = one 16x16 output tile, K=4 -> exactly one v_wmma_f32_16x16x4_f32.
// ---------------------------------------------------------------------------
__global__ __launch_bounds__(256) void embed_wmma(
    const float* __restrict__ x, const float* __restrict__ W1,
    const float* __restrict__ b1, const float* __restrict__ g1,
    const float* __restrict__ beta1, float* __restrict__ h0)
{
    const int lane = threadIdx.x & 31;
    const int wave = threadIdx.x >> 5;
    const int tile = blockIdx.x * 8 + wave;   // 2048 tiles = 512 Mt x 4 Nt
    const int mt = tile >> 2;
    const int nt = tile & 3;
    const int half = lane >> 4;               // 0: lanes 0-15, 1: lanes 16-31
    const int l16  = lane & 15;

    // A fragment: 16x4 f32. VGPR0 = K{0,2}, VGPR1 = K{1,3} split across halves.
    const int m  = mt * 16 + l16;
    const int k0 = half * 2;
    v2f a = *(const v2f*)(x + m * 4 + k0);    // contiguous pair, 8B aligned

    // B fragment: 4x16 f32, same K-split across lane halves.
    const int col = nt * 16 + l16;
    v2f b;
    b.x = W1[(k0 + 0) * DCH + col];
    b.y = W1[(k0 + 1) * DCH + col];

    v8f c = {};
    c = __builtin_amdgcn_wmma_f32_16x16x4_f32(
            /*neg_a=*/false, a, /*neg_b=*/false, b,
            /*c_mod=*/(short)0, c, /*reuse_a=*/false, /*reuse_b=*/false);

    // Epilogue: bias + relu + bn (all depend on column only).
    const float bb = b1[col], gg = g1[col], bt = beta1[col];
    #pragma unroll
    for (int v = 0; v < 8; ++v) {
        const int row = mt * 16 + half * 8 + v;  // C/D layout: VGPR v -> rows v, v+8
        float z = c[v] + bb;
        z = z > 0.f ? z : 0.f;
        h0[row * DCH + col] = gg * z + bt;
    }
}

// ---------------------------------------------------------------------------
// Kernel 2: t = h @ [Wi | Wj]   h:(8192,64)  We:(128,64) -> t:(8192,128)
// t[:, :64] = ti = h @ We[:64],  t[:, 64:] = tj = h @ We[64:].
// One wave owns a 16-row strip and ALL 8 column tiles: 8 f32 accumulators
// (64 VGPRs), A fragment reused by 8 WMMAs per K-step, 128 WMMAs per wave.
// B (32 KB) is staged in LDS once per block and shared by the 8 waves.
// ---------------------------------------------------------------------------
__global__ __launch_bounds__(256) void edge_gemm_wmma(
    const float* __restrict__ h, const float* __restrict__ We,
    float* __restrict__ t)
{
    __shared__ float bs[DCH * BSTRIDE];  // combined B[k][j], k<64, j<128 (padded pitch)

    // Cooperative stage of B = [Wi | Wj] into LDS (8192 floats, 32 per thread).
    for (int i = threadIdx.x; i < DCH * 128; i += 256) {
        const int k = i >> 7;
        const int j = i & 127;
        const float v = (j < DCH) ? We[k * DCH + j]
                                  : We[(DCH + k) * DCH + (j - DCH)];
        bs[k * BSTRIDE + j] = v;
    }
    __syncthreads();

    const int lane = threadIdx.x & 31;
    const int wave = threadIdx.x >> 5;
    const int mt   = blockIdx.x * 8 + wave;   // 512 row tiles / 64 blocks
    const int half = lane >> 4;
    const int l16  = lane & 15;

    const int m = mt * 16 + l16;
    const float* __restrict__ arow = h + m * DCH;

    v8f c[8];
    #pragma unroll
    for (int nt = 0; nt < 8; ++nt) c[nt] = (v8f){};

    #pragma unroll
    for (int kk = 0; kk < 16; ++kk) {
        const int k0 = kk * 4 + half * 2;     // K-split: lanes<16 -> K,K+1; >=16 -> K+2,K+3
        v2f a = *(const v2f*)(arow + k0);     // one b64 load, reused by 8 WMMAs
        const float* __restrict__ brow = bs + k0 * BSTRIDE;
        #pragma unroll
        for (int nt = 0; nt < 8; ++nt) {
            const int colc = nt * 16 + l16;
            v2f b;
            b.x = brow[colc];                 // row k0
            b.y = brow[BSTRIDE + colc];       // row k0+1
            c[nt] = __builtin_amdgcn_wmma_f32_16x16x4_f32(
                        false, a, false, b, (short)0, c[nt], false, false);
        }
    }

    #pragma unroll
    for (int nt = 0; nt < 8; ++nt) {
        const int colc = nt * 16 + l16;
        #pragma unroll
        for (int v = 0; v < 8; ++v) {
            const int row = mt * 16 + half * 8 + v;
            t[row * 128 + colc] = c[nt][v];
        }
    }
}

// ---------------------------------------------------------------------------
// Kernel 3: EdgeConv collapse. Per batch b, per channel d:
//   sel_d = (ge_d >= 0 ? max_n tj[b,n,d] : min_n tj[b,n,d]) + be_d
//   hout[b,n,d] = ge_d * relu(ti[b,n,d] + sel_d) + betae_d
// Exact: max_m(g*relu(ti+tj_m+b))+beta commutes through monotone relu/scale.
// Block = one batch, 256 threads = 4 n-chunks x 64 channels.
// ---------------------------------------------------------------------------
__global__ __launch_bounds__(256) void edge_reduce_combine(
    const float* __restrict__ t, const float* __restrict__ be,
    const float* __restrict__ ge, const float* __restrict__ betae,
    float* __restrict__ hout)
{
    __shared__ float smax[4][DCH];
    __shared__ float smin[4][DCH];
    __shared__ float ssel[DCH];

    const int b     = blockIdx.x;
    const int d     = threadIdx.x & 63;
    const int chunk = threadIdx.x >> 6;

    const float* __restrict__ tj = t + (size_t)b * NPTS * 128 + DCH;
    float mx = -3.402823466e38f, mn = 3.402823466e38f;
    for (int n = chunk * 64; n < chunk * 64 + 64; ++n) {
        const float v = tj[n * 128 + d];
        mx = fmaxf(mx, v);
        mn = fminf(mn, v);
    }
    smax[chunk][d] = mx;
    smin[chunk][d] = mn;
    __syncthreads();

    if (threadIdx.x < DCH) {
        const float M  = fmaxf(fmaxf(smax[0][d], smax[1][d]),
                               fmaxf(smax[2][d], smax[3][d]));
        const float m2 = fminf(fminf(smin[0][d], smin[1][d]),
                               fminf(smin[2][d], smin[3][d]));
        ssel[d] = ((ge[d] >= 0.f) ? M : m2) + be[d];
    }
    __syncthreads();

    const float* __restrict__ ti = t + (size_t)b * NPTS * 128;
    const float gg = ge[d], bt = betae[d], sel = ssel[d];
    for (int n = chunk * 64; n < chunk * 64 + 64; ++n) {
        float z = ti[n * 128 + d] + sel;
        z = z > 0.f ? z : 0.f;
        hout[((size_t)b * NPTS + n) * DCH + d] = gg * z + bt;
    }
}

// ---------------------------------------------------------------------------
// Kernel 4: global mean/max pool + MLP head. Block = one batch.
// ---------------------------------------------------------------------------
__global__ __launch_bounds__(128) void head_kernel(
    const float* __restrict__ h, const float* __restrict__ Wg1,
    const float* __restrict__ bg1, const float* __restrict__ Wg2,
    const float* __restrict__ bg2, float* __restrict__ out)
{
    __shared__ float xg[2 * DCH];
    __shared__ float y[DCH];
    const int b  = blockIdx.x;
    const int tt = threadIdx.x;

    if (tt < DCH) {
        const float* __restrict__ hb = h + (size_t)b * NPTS * DCH + tt;
        float s = 0.f, mx = -3.402823466e38f;
        for (int n = 0; n < NPTS; ++n) {
            const float v = hb[n * DCH];
            s += v;
            mx = fmaxf(mx, v);
        }
        xg[tt]       = s * (1.0f / NPTS);  // mean
        xg[DCH + tt] = mx;                 // max
    }
    __syncthreads();

    if (tt < DCH) {
        float acc = bg1[tt];
        for (int k = 0; k < 2 * DCH; ++k) acc += xg[k] * Wg1[k * DCH + tt];
        y[tt] = acc > 0.f ? acc : 0.f;
    }
    __syncthreads();

    if (tt < 2) {
        float acc = bg2[tt];
        for (int e = 0; e < DCH; ++e) acc += y[e] * Wg2[e * 2 + tt];
        out[b * 2 + tt] = acc;
    }
}

// ---------------------------------------------------------------------------
extern "C" void kernel_launch(void* const* d_in, const int* in_sizes, int n_in,
                              void* d_out, int out_size, void* d_ws, size_t ws_size,
                              hipStream_t stream) {
    (void)in_sizes; (void)n_in; (void)out_size; (void)ws_size;
    const float* x      = (const float*)d_in[0];
    const float* W1     = (const float*)d_in[1];
    const float* b1     = (const float*)d_in[2];
    const float* g1     = (const float*)d_in[3];
    const float* beta1  = (const float*)d_in[4];
    const float* We1    = (const float*)d_in[5];
    const float* be1    = (const float*)d_in[6];
    const float* ge1    = (const float*)d_in[7];
    const float* betae1 = (const float*)d_in[8];
    const float* We2    = (const float*)d_in[9];
    const float* be2    = (const float*)d_in[10];
    const float* ge2    = (const float*)d_in[11];
    const float* betae2 = (const float*)d_in[12];
    const float* Wg1    = (const float*)d_in[13];
    const float* bg1    = (const float*)d_in[14];
    const float* Wg2    = (const float*)d_in[15];
    const float* bg2    = (const float*)d_in[16];
    float* out = (float*)d_out;

    float* hbuf = (float*)d_ws;           // 8192*64 floats (reused for h0,h1,h2)
    float* tbuf = hbuf + NROWS * DCH;     // 8192*128 floats (ti|tj, reused)

    embed_wmma         <<<256, 256, 0, stream>>>(x, W1, b1, g1, beta1, hbuf);
    edge_gemm_wmma     <<< 64, 256, 0, stream>>>(hbuf, We1, tbuf);
    edge_reduce_combine<<< 32, 256, 0, stream>>>(tbuf, be1, ge1, betae1, hbuf);
    edge_gemm_wmma     <<< 64, 256, 0, stream>>>(hbuf, We2, tbuf);
    edge_reduce_combine<<< 32, 256, 0, stream>>>(tbuf, be2, ge2, betae2, hbuf);
    head_kernel        <<< 32, 128, 0, stream>>>(hbuf, Wg1, bg1, Wg2, bg2, out);
}